// TimelinePrototypeMatcher_38362647888273
// MI455X (gfx1250) — compile-verified
//
#include <hip/hip_runtime.h>
#include <math.h>

typedef float v2f __attribute__((ext_vector_type(2)));
typedef float v8f __attribute__((ext_vector_type(8)));

#define B_   32
#define T_   512
#define D_   512
#define P_   200
#define NP_  8
#define NEGV (-100000.0f)
#define EPSN 1e-12f
#define LSTR 18   // LDS row stride (floats) for 512x16 dist tile, conflict-avoiding

// ---------------------------------------------------------------------------
// Kernel 1: L2-normalize x rows (B*T rows of D floats). One wave per row.
// ---------------------------------------------------------------------------
__global__ __launch_bounds__(256) void norm_x_kernel(const float* __restrict__ x,
                                                     float* __restrict__ xn) {
  int row  = blockIdx.x * 8 + (threadIdx.x >> 5);
  int lane = threadIdx.x & 31;
  const float* src = x + (size_t)row * D_;
  float v[16];
  float ss = 0.f;
#pragma unroll
  for (int j = 0; j < 16; ++j) { v[j] = src[lane + 32 * j]; ss += v[j] * v[j]; }
#pragma unroll
  for (int off = 16; off > 0; off >>= 1) ss += __shfl_xor(ss, off, 32);
  float inv = 1.f / fmaxf(sqrtf(ss), EPSN);
  float* dst = xn + (size_t)row * D_;
#pragma unroll
  for (int j = 0; j < 16; ++j) dst[lane + 32 * j] = v[j] * inv;
}

// ---------------------------------------------------------------------------
// Kernel 2: normalize prototypes over D (axis=1) and transpose to
// pnT[n = p*NP + i][d], row-major over d. One wave per (p,i).
// ---------------------------------------------------------------------------
__global__ __launch_bounds__(256) void norm_p_kernel(const float* __restrict__ pv,
                                                     float* __restrict__ pnT) {
  int idx  = blockIdx.x * 8 + (threadIdx.x >> 5);   // 0..1599 = p*8+i
  int lane = threadIdx.x & 31;
  int p = idx >> 3, i = idx & 7;
  float v[16];
  float ss = 0.f;
#pragma unroll
  for (int j = 0; j < 16; ++j) {
    int d = lane + 32 * j;
    v[j] = pv[((size_t)p * D_ + d) * NP_ + i];
    ss += v[j] * v[j];
  }
#pragma unroll
  for (int off = 16; off > 0; off >>= 1) ss += __shfl_xor(ss, off, 32);
  float inv = 1.f / fmaxf(sqrtf(ss), EPSN);
  float* dst = pnT + (size_t)idx * D_;
#pragma unroll
  for (int j = 0; j < 16; ++j) dst[lane + 32 * j] = v[j] * inv;
}

// ---------------------------------------------------------------------------
// Kernel 3: fused GEMM (WMMA f32 16x16x4) + greedy selection + scoring.
// grid = (P/2, B); block = 256 (8 waves). Each block: one batch b, two
// prototypes p0, p0+1 -> one 512x16 dist tile in LDS.
// ---------------------------------------------------------------------------
__global__ __launch_bounds__(256) void match_kernel(const float* __restrict__ xn,
                                                    const float* __restrict__ pnT,
                                                    const float* __restrict__ tmask,
                                                    const float* __restrict__ psel,
                                                    float* __restrict__ out) {
  __shared__ float dist[T_ * LSTR];   // 512*18*4 = 36 KB

  const int b  = blockIdx.y;
  const int p0 = blockIdx.x * 2;
  const int wave = threadIdx.x >> 5;
  const int lane = threadIdx.x & 31;
  const int lo = lane & 15;      // M or N index within tile
  const int hi = lane >> 4;      // K-half selector

  // ---- GEMM phase: each wave owns t rows [wave*64, wave*64+64) ----
  const int tb = wave * 64;
  const float* Abase = xn  + ((size_t)(b * T_ + tb + lo)) * D_ + 2 * hi;
  const float* Bbase = pnT + ((size_t)(p0 * NP_ + lo))    * D_ + 2 * hi;

  v8f acc0 = {}, acc1 = {}, acc2 = {}, acc3 = {};
#pragma unroll 4
  for (int k = 0; k < D_; k += 4) {
    v2f bf = *(const v2f*)(Bbase + k);
    v2f a0 = *(const v2f*)(Abase + k);
    v2f a1 = *(const v2f*)(Abase + (size_t)16 * D_ + k);
    v2f a2 = *(const v2f*)(Abase + (size_t)32 * D_ + k);
    v2f a3 = *(const v2f*)(Abase + (size_t)48 * D_ + k);
    acc0 = __builtin_amdgcn_wmma_f32_16x16x4_f32(false, a0, false, bf, (short)0, acc0, false, false);
    acc1 = __builtin_amdgcn_wmma_f32_16x16x4_f32(false, a1, false, bf, (short)0, acc1, false, false);
    acc2 = __builtin_amdgcn_wmma_f32_16x16x4_f32(false, a2, false, bf, (short)0, acc2, false, false);
    acc3 = __builtin_amdgcn_wmma_f32_16x16x4_f32(false, a3, false, bf, (short)0, acc3, false, false);
  }

  // Apply timeline mask, scatter D tiles to LDS: lane->(M = r+8*hi, N = lo)
  const float* tmrow = tmask + (size_t)b * T_;
  {
    v8f accs[4] = {acc0, acc1, acc2, acc3};
#pragma unroll
    for (int m = 0; m < 4; ++m) {
#pragma unroll
      for (int r = 0; r < 8; ++r) {
        int t = tb + m * 16 + r + 8 * hi;
        float tm = tmrow[t];
        dist[t * LSTR + lo] = accs[m][r] * tm + (1.f - tm) * NEGV;
      }
    }
  }
  __syncthreads();

  // ---- Selection phase: waves 0 and 1 each handle one prototype ----
  if (wave >= 2) return;
  const int p  = p0 + wave;
  const int cb = wave * 8;          // column base in LDS tile
  const int i  = lane >> 2;         // subpatch column this lane scans
  const int q  = lane & 3;          // t residue class this lane scans

  unsigned mact[4] = {0xFFFFFFFFu, 0xFFFFFFFFu, 0xFFFFFFFFu, 0xFFFFFFFFu};
  unsigned usedSub = 0;
  int   subs[NP_], evs[NP_];
  float vals[NP_];
  int ev_prev = 0;

  for (int it = 0; it < NP_; ++it) {
    float bv = -3.4e38f;
    int   bt = 0;
    const bool subOK = ((usedSub >> i) & 1u) == 0u;
    for (int s = 0; s < 128; ++s) {
      int t = (s << 2) | q;
      float d = dist[t * LSTR + cb + i];
      bool ma = (mact[s >> 5] >> (s & 31)) & 1u;
      bool adj;
      if (it == 0) adj = true;
      else {
        int dt = t - ev_prev;
        adj = (it == 1) ? (dt >= -3 && dt <= 3) : (dt >= 1 && dt <= 3);
      }
      float dm = d + ((ma && subOK && adj) ? 0.f : NEGV);
      if (dm > bv || (dm == bv && t < bt)) { bv = dm; bt = t; }
    }
    // reduce over q (within column group of 4): max val, tie -> min t
#pragma unroll
    for (int off = 1; off < 4; off <<= 1) {
      float ov = __shfl_xor(bv, off, 32);
      int   ot = __shfl_xor(bt, off, 32);
      if (ov > bv || (ov == bv && ot < bt)) { bv = ov; bt = ot; }
    }
    // reduce over i (8 groups): max val, tie -> min i (carry t along)
    int bi = i;
#pragma unroll
    for (int off = 4; off < 32; off <<= 1) {
      float ov = __shfl_xor(bv, off, 32);
      int   ot = __shfl_xor(bt, off, 32);
      int   oi = __shfl_xor(bi, off, 32);
      if (ov > bv || (ov == bv && oi < bi)) { bv = ov; bt = ot; bi = oi; }
    }
    // all lanes now agree on (bv, bt, bi); update state
    vals[it] = bv; evs[it] = bt; subs[it] = bi;
    if ((bt & 3) == q) { int s0 = bt >> 2; mact[s0 >> 5] &= ~(1u << (s0 & 31)); }
    usedSub |= (1u << bi);
    ev_prev = bt;
  }

  if (lane == 0) {
    // stable ascending argsort of subs (ties keep iteration order)
    int ord[NP_];
    bool taken[NP_] = {false, false, false, false, false, false, false, false};
    for (int r = 0; r < NP_; ++r) {
      int bestj = -1;
      for (int j = 0; j < NP_; ++j) {
        if (taken[j]) continue;
        if (bestj < 0 || subs[j] < subs[bestj]) bestj = j;
      }
      taken[bestj] = true;
      ord[r] = bestj;
    }
    float slots[NP_], fs = 0.f;
#pragma unroll
    for (int s = 0; s < NP_; ++s) {
      float xv = psel[p * NP_ + s];
      float sg = 1.f / (1.f + expf(-xv));
      slots[s] = sg; fs += sg;
    }
    fs += 1e-10f;
    float mas = 0.f;
#pragma unroll
    for (int s = 0; s < NP_; ++s) mas += vals[ord[s]] * slots[s] * (float)NP_ / fs;

    const int bp = b * P_ + p;
    out[bp] = mas;                       // max_activation_slots
    out[B_ * P_ + bp] = (float)NP_ - mas; // min_distances
    float* idx_out = out + 2 * B_ * P_ + (size_t)bp * NP_;
#pragma unroll
    for (int s = 0; s < NP_; ++s) idx_out[s] = (float)evs[ord[s]];
  }
}

// ---------------------------------------------------------------------------
extern "C" void kernel_launch(void* const* d_in, const int* in_sizes, int n_in,
                              void* d_out, int out_size, void* d_ws, size_t ws_size,
                              hipStream_t stream) {
  const float* x     = (const float*)d_in[0];  // (B,T,D)
  const float* tmask = (const float*)d_in[1];  // (B,T)
  const float* pv    = (const float*)d_in[2];  // (P,D,NP)
  const float* psel  = (const float*)d_in[3];  // (1,P,NP)
  float* out = (float*)d_out;

  float* xn  = (float*)d_ws;                         // (B*T, D) = 33.5 MB
  float* pnT = xn + (size_t)B_ * T_ * D_;            // (P*NP, D) = 3.2 MB

  norm_x_kernel<<<(B_ * T_) / 8, 256, 0, stream>>>(x, xn);
  norm_p_kernel<<<(P_ * NP_) / 8, 256, 0, stream>>>(pv, pnT);

  dim3 grid(P_ / 2, B_);
  match_kernel<<<grid, 256, 0, stream>>>(xn, pnT, tmask, psel, out);
}